// WeightNet_17806934409816
// MI455X (gfx1250) — compile-verified
//
#include <hip/hip_runtime.h>

typedef __attribute__((ext_vector_type(2))) float v2f;
typedef __attribute__((ext_vector_type(8))) float v8f;

#define NB    32
#define INP   64
#define OUP   64
#define HH    112
#define WW    112
#define HWSZ  (HH * WW)          // 12544
#define WPC   (INP * 9)          // 576 weights per output channel
#define WPB   (OUP * WPC)        // 36864 weights per batch sample

// ---------------------------------------------------------------------------
// Stage 1 (tiny): h = sigmoid(gap @ w1^T + b1); xw[b,t] = sum_i h[b,(t/144)*4+i]*w2[t,i]
// One block per batch sample. Output: per-sample conv weights in workspace,
// layout [b][co][ci][ky][kx] (row-major, matches reference reshape).
// ---------------------------------------------------------------------------
__global__ __launch_bounds__(256) void wn_gen_weights(
    const float* __restrict__ x_gap, const float* __restrict__ w1,
    const float* __restrict__ b1,    const float* __restrict__ w2,
    float* __restrict__ wout) {
  __shared__ float gsh[16];
  __shared__ float hsh[1024];
  const int b   = blockIdx.x;
  const int tid = threadIdx.x;

  if (tid < 16) gsh[tid] = x_gap[b * 16 + tid];
  __syncthreads();

  // each thread computes 4 of the 1024 h values
  #pragma unroll
  for (int q = 0; q < 4; ++q) {
    const int j = tid * 4 + q;
    float acc = b1[j];
    const float* wr = w1 + j * 16;
    #pragma unroll
    for (int i = 0; i < 16; ++i) acc += gsh[i] * wr[i];
    hsh[j] = 1.0f / (1.0f + __expf(-acc));
  }
  __syncthreads();

  float* wb = wout + (size_t)b * WPB;
  for (int t = tid; t < WPB; t += 256) {
    const int g = t / 144;                 // group index 0..255
    const float* hr  = hsh + g * 4;
    const float* w2r = w2 + (size_t)t * 4; // w2 rows are contiguous (t, i)
    wb[t] = hr[0] * w2r[0] + hr[1] * w2r[1] + hr[2] * w2r[2] + hr[3] * w2r[3];
  }
}

// ---------------------------------------------------------------------------
// Stage 2 (heavy): per-sample 64->64 3x3 conv as implicit GEMM with
// V_WMMA_F32_16X16X4_F32.  grid = (pixel_block 0..48, co_tile 0..3, b 0..31),
// 256 threads = 8 waves; each wave computes a 16(co) x 32(pixel) tile as two
// 16x16 sub-tiles sharing A-fragments (halves LDS reads per WMMA).
// ---------------------------------------------------------------------------
__global__ __launch_bounds__(256) void wn_conv(
    const float* __restrict__ x, const float* __restrict__ wgt,
    float* __restrict__ out) {
  // LDS weight slab, transposed to [u = ci*9+kk][co16] for conflict-free A reads
  __shared__ float wl[WPC * 16];

  const int pb  = blockIdx.x;   // 256-pixel block
  const int ct  = blockIdx.y;   // co tile (16 channels)
  const int b   = blockIdx.z;
  const int tid = threadIdx.x;

  // cooperative stage of 16co x 576 weights: global [co][u] -> lds [u][co]
  const float* wg = wgt + (size_t)b * WPB + (size_t)ct * 16 * WPC;
  for (int i = tid; i < 16 * WPC; i += 256) {
    const int co = i / WPC;
    const int u  = i - co * WPC;
    wl[u * 16 + co] = wg[i];
  }
  __syncthreads();

  const int wid  = tid >> 5;
  const int lane = tid & 31;
  const int n    = lane & 15;          // N index (pixel) == M index (co) for A
  const int kb   = (lane >> 4) << 1;   // K base: 0 (lanes 0-15) / 2 (lanes 16-31)

  // two 16-pixel sub-tiles; each lies within one image row (112 % 16 == 0)
  const int p0  = pb * 256 + wid * 32;
  const int p1  = p0 + 16;
  const int y0  = p0 / WW;
  const int x00 = p0 - y0 * WW;
  const int y1  = p1 / WW;
  const int x01 = p1 - y1 * WW;

  const float* xb = x + (size_t)b * INP * HWSZ;

  v8f c0 = {0.f, 0.f, 0.f, 0.f, 0.f, 0.f, 0.f, 0.f};
  v8f c1 = {0.f, 0.f, 0.f, 0.f, 0.f, 0.f, 0.f, 0.f};

  #pragma unroll
  for (int kk = 0; kk < 9; ++kk) {
    const int ky = kk / 3;
    const int kx = kk - ky * 3;

    // sub-tile 0 coordinates / halo mask
    const int  yy0 = y0 + ky - 1;
    const int  xx0 = x00 + n + kx - 1;
    const float m0 = ((yy0 >= 0) & (yy0 < HH) & (xx0 >= 0) & (xx0 < WW)) ? 1.0f : 0.0f;
    const int  pix0 = min(max(yy0, 0), HH - 1) * WW + min(max(xx0, 0), WW - 1);

    // sub-tile 1 coordinates / halo mask
    const int  yy1 = y1 + ky - 1;
    const int  xx1 = x01 + n + kx - 1;
    const float m1 = ((yy1 >= 0) & (yy1 < HH) & (xx1 >= 0) & (xx1 < WW)) ? 1.0f : 0.0f;
    const int  pix1 = min(max(yy1, 0), HH - 1) * WW + min(max(xx1, 0), WW - 1);

    #pragma unroll 4
    for (int cq = 0; cq < 16; ++cq) {
      const int ci0 = cq * 4 + kb;         // this lane's K pair: ci0, ci0+1
      const float* p = xb + (size_t)ci0 * HWSZ;

      // B fragments (masked for halo padding)
      const float b00 = p[pix0]        * m0;
      const float b10 = p[pix0 + HWSZ] * m0;
      const float b01 = p[pix1]        * m1;
      const float b11 = p[pix1 + HWSZ] * m1;

      // shared A fragment from LDS: W[co=n, ci0/ci0+1, kk]
      v2f af = { wl[(ci0 * 9 + kk) * 16 + n],
                 wl[((ci0 + 1) * 9 + kk) * 16 + n] };
      v2f bf0 = { b00, b10 };
      v2f bf1 = { b01, b11 };

      c0 = __builtin_amdgcn_wmma_f32_16x16x4_f32(
               false, af, false, bf0, (short)0, c0, false, false);
      c1 = __builtin_amdgcn_wmma_f32_16x16x4_f32(
               false, af, false, bf1, (short)0, c1, false, false);
    }
  }

  // Store D: lane n = N (pixel), VGPR r -> M = r + (lane>=16 ? 8 : 0) (co)
  float* ob0 = out + (size_t)b * OUP * HWSZ + (size_t)(ct * 16) * HWSZ + p0 + n;
  float* ob1 = ob0 + 16;
  const int mbase = (lane >> 4) * 8;
  #pragma unroll
  for (int r = 0; r < 8; ++r) {
    ob0[(size_t)(mbase + r) * HWSZ] = c0[r];
    ob1[(size_t)(mbase + r) * HWSZ] = c1[r];
  }
}

// ---------------------------------------------------------------------------
extern "C" void kernel_launch(void* const* d_in, const int* in_sizes, int n_in,
                              void* d_out, int out_size, void* d_ws, size_t ws_size,
                              hipStream_t stream) {
  const float* x    = (const float*)d_in[0];
  const float* xgap = (const float*)d_in[1];
  const float* w1   = (const float*)d_in[2];
  const float* b1   = (const float*)d_in[3];
  const float* w2   = (const float*)d_in[4];
  float* out = (float*)d_out;
  float* wsc = (float*)d_ws;   // 32*36864 floats = 4.7 MB of generated weights

  wn_gen_weights<<<dim3(NB), dim3(256), 0, stream>>>(xgap, w1, b1, w2, wsc);
  wn_conv<<<dim3(HWSZ / 256, OUP / 16, NB), dim3(256), 0, stream>>>(x, wsc, out);
}